// MultiModelNetV2_49744311222536
// MI455X (gfx1250) — compile-verified
//
#include <hip/hip_runtime.h>
#include <math.h>

// ---------------------------------------------------------------------------
// Problem constants (must match reference)
// ---------------------------------------------------------------------------
#define HH      4
#define DD      32
#define EMBD    128
#define EE      200000
#define NT_ALL  125000
#define INV_SQRT_D 0.17677669529663687f

static const int CNT[4]      = {30000, 20000, 25000, 50000};
static const int NODE_OFF[4] = {0, 30000, 50000, 75000};
static const int ES[8]       = {0,1,2,1,3,0,3,2};
static const int ET[8]       = {1,0,1,2,0,3,2,3};
static const int OUT_OFF[3]  = {0, 30000, 50000};

typedef float v2f __attribute__((ext_vector_type(2)));
typedef float v8f __attribute__((ext_vector_type(8)));

#define ACT_NONE 0
#define ACT_RELU 1

// ---------------------------------------------------------------------------
// fp32 WMMA GEMM:  C[m,n] (=|+=) act( sum_k A[m,k]*B(n,k) + bias[n] )
//   KT  : compile-time K (32/64/128) -> fully unrolled, immediate offsets
//   MTL : M-tiles per wave (each 16 rows) sharing the B fragments
//   NTL : N-tiles per wave (each 16 cols) sharing the A fragments
//   BKMAJ/LDBT : B stored K-major with compile-time ldb (m_rel path)
// Per K-step: MTL+NTL b64 loads feed MTL*NTL WMMAs (2+4 -> 8).
// Requirements: N % (16*NTL) == 0 (all call sites: 32, 128, 384).
// Rows are clamped (not predicated): out-of-range rows compute garbage that is
// never stored; each C element depends only on its own A row / B column.
// A-frag (ISA 7.12.2, 32-bit A 16x4): lanes 0-15 rows, VGPRs = K,K+1;
// lanes 16-31 same rows, K+2,K+3.  B mirrors with N in place of M.
// C: VGPR v -> row v (lanes 0-15) / v+8 (lanes 16-31), col = lane%16.
// ---------------------------------------------------------------------------
template<int KT, int MTL, int NTL, bool BKMAJ, int LDBT, int ACT, bool ACC>
__global__ __launch_bounds__(32)
void wmma_gemm(float* __restrict__ C, const float* __restrict__ A,
               const float* __restrict__ B, const float* __restrict__ bias,
               int M, int lda, int ldb, int ldc)
{
    const int lane = threadIdx.x & 31;
    const int half = lane >> 4;          // 0 or 1  (selects K pair)
    const int l16  = lane & 15;
    const int colbase = blockIdx.x * (16 * NTL);
    const int rowbase = blockIdx.y * (16 * MTL);

    const float* Ar[MTL];
#pragma unroll
    for (int m = 0; m < MTL; ++m) {
        int arow = rowbase + 16 * m + l16;
        int arow_c = (arow < M) ? arow : (M - 1);      // clamp, no predication
        Ar[m] = A + (size_t)arow_c * lda + half * 2;
    }

    const float* Bp[NTL];
#pragma unroll
    for (int i = 0; i < NTL; ++i) {
        const int col = colbase + 16 * i + l16;
        if (BKMAJ) Bp[i] = B + col + (size_t)(half * 2) * LDBT;
        else       Bp[i] = B + (size_t)col * ldb + half * 2;
    }

    v8f acc[MTL][NTL];
#pragma unroll
    for (int m = 0; m < MTL; ++m)
#pragma unroll
        for (int i = 0; i < NTL; ++i)
            acc[m][i] = (v8f){0.f,0.f,0.f,0.f,0.f,0.f,0.f,0.f};

#pragma unroll
    for (int k = 0; k < KT; k += 4) {
        v2f a[MTL];
#pragma unroll
        for (int m = 0; m < MTL; ++m) a[m] = *(const v2f*)(Ar[m] + k);
        v2f b[NTL];
#pragma unroll
        for (int i = 0; i < NTL; ++i) {
            if (BKMAJ) {
                b[i].x = Bp[i][(size_t)k * LDBT];
                b[i].y = Bp[i][(size_t)k * LDBT + LDBT];
            } else {
                b[i] = *(const v2f*)(Bp[i] + k);
            }
        }
#pragma unroll
        for (int m = 0; m < MTL; ++m)
#pragma unroll
            for (int i = 0; i < NTL; ++i)
                acc[m][i] = __builtin_amdgcn_wmma_f32_16x16x4_f32(
                                false, a[m], false, b[i], (short)0, acc[m][i], false, false);
    }

#pragma unroll
    for (int m = 0; m < MTL; ++m) {
        const int rbase = rowbase + 16 * m + half * 8;
#pragma unroll
        for (int i = 0; i < NTL; ++i) {
            const int col = colbase + 16 * i + l16;
            const float bs = (bias != nullptr) ? bias[col] : 0.f;
#pragma unroll
            for (int v = 0; v < 8; ++v) {
                const int r = rbase + v;
                if (r < M) {
                    float val = acc[m][i][v] + bs;
                    if (ACT == ACT_RELU) val = fmaxf(val, 0.f);
                    if (ACC) C[(size_t)r * ldc + col] += val;
                    else     C[(size_t)r * ldc + col]  = val;
                }
            }
        }
    }
}

// ---------------------------------------------------------------------------
// Elementwise / edge kernels
// ---------------------------------------------------------------------------
__global__ void fill_kernel(float* __restrict__ p, long n, float v)
{
    long i = (long)blockIdx.x * blockDim.x + threadIdx.x;
    if (i < n) p[i] = v;
}

__device__ inline void atomicMaxFloat(float* addr, float v)
{
    if (v >= 0.f) atomicMax((int*)addr, __float_as_int(v));
    else          atomicMin((unsigned int*)addr, (unsigned int)__float_as_int(v));
}

// logit[e,h] = (K[src] . qa[dst])_h * p_rel[h] * inv_sqrt_d ; atomicMax into MX
__global__ void logit_kernel(float* __restrict__ LG, float* __restrict__ MX,
                             const float* __restrict__ KB, const float* __restrict__ QA,
                             const int* __restrict__ src, const int* __restrict__ dst,
                             const float* __restrict__ prel, int offS, int offT)
{
    int idx = blockIdx.x * blockDim.x + threadIdx.x;
    if (idx >= EE * HH) return;
    int e = idx >> 2, h = idx & 3;
    int s = src[e], t = dst[e];
    const float4* kp = (const float4*)(KB + (size_t)(offS + s) * EMBD + h * DD);
    const float4* qp = (const float4*)(QA + (size_t)t * EMBD + h * DD);
    float acc = 0.f;
#pragma unroll
    for (int d = 0; d < 8; ++d) {
        float4 a = kp[d], b = qp[d];
        acc += a.x * b.x + a.y * b.y + a.z * b.z + a.w * b.w;
    }
    acc *= prel[h] * INV_SQRT_D;
    LG[e * HH + h] = acc;
    atomicMaxFloat(&MX[(size_t)(offT + t) * HH + h], acc);
}

// ex = exp(logit - mx); DEN += ex; TMP[dst,f] += ex * V[src,f]
__global__ void edge_accum_kernel(float* __restrict__ TMPb, float* __restrict__ DEN,
                                  const float* __restrict__ LG, const float* __restrict__ MX,
                                  const float* __restrict__ VB,
                                  const int* __restrict__ src, const int* __restrict__ dst,
                                  int offS, int offT)
{
    long idx = (long)blockIdx.x * blockDim.x + threadIdx.x;
    if (idx >= (long)EE * EMBD) return;
    int e = (int)(idx >> 7);
    int f = (int)(idx & 127);
    int h = f >> 5;
    int s = src[e], t = dst[e];
    float ex = expf(LG[e * HH + h] - MX[(size_t)(offT + t) * HH + h]);
    if ((f & 31) == 0) atomicAdd(&DEN[(size_t)(offT + t) * HH + h], ex);
    atomicAdd(&TMPb[(size_t)t * EMBD + f], ex * VB[(size_t)(offS + s) * EMBD + f]);
}

// G = gelu( AGG / max(DEN,1e-16) )   (exact gelu)
__global__ void gelu_div_kernel(float* __restrict__ G, const float* __restrict__ AGGt,
                                const float* __restrict__ DENt, int N)
{
    long idx = (long)blockIdx.x * blockDim.x + threadIdx.x;
    if (idx >= (long)N * EMBD) return;
    int n = (int)(idx >> 7), c = (int)(idx & 127);
    float den = fmaxf(DENt[(size_t)n * HH + (c >> 5)], 1e-16f);
    float g = AGGt[(size_t)n * EMBD + c] / den;
    G[idx] = 0.5f * g * (1.f + erff(g * 0.70710678118654752f));
}

// o = sg*o + (1-sg)*x ; y = relu(LN(o + x))
__global__ void node_fin_kernel(float* __restrict__ dst, const float* __restrict__ O,
                                const float* __restrict__ Xin, const float* __restrict__ skipPtr,
                                const float* __restrict__ g, const float* __restrict__ b)
{
    int n = blockIdx.x, c = threadIdx.x;
    float o  = O[(size_t)n * EMBD + c];
    float x  = Xin[(size_t)n * EMBD + c];
    float sg = 1.f / (1.f + expf(-skipPtr[0]));
    float y  = sg * o + (1.f - sg) * x + x;

    __shared__ float ws[4];
    float v = y;
    for (int off = 16; off; off >>= 1) v += __shfl_xor(v, off, 32);
    if ((c & 31) == 0) ws[c >> 5] = v;
    __syncthreads();
    float mean = (ws[0] + ws[1] + ws[2] + ws[3]) * (1.f / 128.f);
    __syncthreads();
    float d  = y - mean;
    float v2 = d * d;
    for (int off = 16; off; off >>= 1) v2 += __shfl_xor(v2, off, 32);
    if ((c & 31) == 0) ws[c >> 5] = v2;
    __syncthreads();
    float var = (ws[0] + ws[1] + ws[2] + ws[3]) * (1.f / 128.f);
    float out = d * rsqrtf(var + 1e-5f) * g[c] + b[c];
    dst[(size_t)n * EMBD + c] = fmaxf(out, 0.f);
}

// SEQ[(n*3+i)*128+c] = L_i[(off+n)*128+c]
__global__ void seq_pack_kernel(float* __restrict__ SEQ, const float* __restrict__ L0,
                                const float* __restrict__ L1, const float* __restrict__ L2,
                                int off, int N)
{
    long idx = (long)blockIdx.x * blockDim.x + threadIdx.x;
    if (idx >= (long)N * 3 * EMBD) return;
    int n = (int)(idx / (3 * EMBD));
    int rem = (int)(idx % (3 * EMBD));
    int i = rem >> 7, c = rem & 127;
    const float* L = (i == 0) ? L0 : ((i == 1) ? L1 : L2);
    SEQ[idx] = L[(size_t)(off + n) * EMBD + c];
}

// 3-token per-node self-attention (H heads of D)
__global__ void fusion_attn_kernel(float* __restrict__ ATT, const float* __restrict__ QKV)
{
    int n = blockIdx.x, c = threadIdx.x;
    int h = c >> 5, d = c & 31;
    const float* base = QKV + (size_t)n * 3 * 384;
#pragma unroll
    for (int i = 0; i < 3; ++i) {
        float lg[3];
#pragma unroll
        for (int j = 0; j < 3; ++j) {
            const float4* qp = (const float4*)(base + i * 384 + h * DD);
            const float4* kp = (const float4*)(base + j * 384 + 128 + h * DD);
            float s = 0.f;
#pragma unroll
            for (int dd = 0; dd < 8; ++dd) {
                float4 a = qp[dd], b = kp[dd];
                s += a.x * b.x + a.y * b.y + a.z * b.z + a.w * b.w;
            }
            lg[j] = s * INV_SQRT_D;
        }
        float mx = fmaxf(lg[0], fmaxf(lg[1], lg[2]));
        float e0 = expf(lg[0] - mx), e1 = expf(lg[1] - mx), e2 = expf(lg[2] - mx);
        float inv = 1.f / (e0 + e1 + e2);
        float o = (e0 * base[0 * 384 + 256 + h * DD + d] +
                   e1 * base[1 * 384 + 256 + h * DD + d] +
                   e2 * base[2 * 384 + 256 + h * DD + d]) * inv;
        ATT[((size_t)n * 3 + i) * EMBD + c] = o;
    }
}

// mean over 3 tokens + LayerNorm (no relu)
__global__ void fusion_final_kernel(float* __restrict__ out, const float* __restrict__ OP,
                                    const float* __restrict__ g, const float* __restrict__ b)
{
    int n = blockIdx.x, c = threadIdx.x;
    const float* p = OP + (size_t)n * 3 * EMBD;
    float y = (p[c] + p[EMBD + c] + p[2 * EMBD + c]) * (1.f / 3.f);

    __shared__ float ws[4];
    float v = y;
    for (int off = 16; off; off >>= 1) v += __shfl_xor(v, off, 32);
    if ((c & 31) == 0) ws[c >> 5] = v;
    __syncthreads();
    float mean = (ws[0] + ws[1] + ws[2] + ws[3]) * (1.f / 128.f);
    __syncthreads();
    float d  = y - mean;
    float v2 = d * d;
    for (int off = 16; off; off >>= 1) v2 += __shfl_xor(v2, off, 32);
    if ((c & 31) == 0) ws[c >> 5] = v2;
    __syncthreads();
    float var = (ws[0] + ws[1] + ws[2] + ws[3]) * (1.f / 128.f);
    out[(size_t)n * EMBD + c] = d * rsqrtf(var + 1e-5f) * g[c] + b[c];
}

// ---------------------------------------------------------------------------
// Host-side GEMM launch helpers (one per compile-time shape)
// ---------------------------------------------------------------------------
// input projection: K=64, N=128, relu; wave computes 32x64 block
static inline void gemm_proj(hipStream_t st, float* C, const float* A, const float* B,
                             const float* bias, int M)
{
    dim3 grid(128 / 64, (M + 31) / 32), blk(32);
    wmma_gemm<64, 2, 4, false, 0, ACT_RELU, false><<<grid, blk, 0, st>>>(C, A, B, bias, M, 64, 64, 128);
}
// dense K=128 GEMM (N multiple of 64): kqv / out-proj / fusion projections
static inline void gemm_k128(hipStream_t st, float* C, const float* A, const float* B,
                             const float* bias, int M, int N, int lda, int ldb, int ldc)
{
    dim3 grid(N / 64, (M + 31) / 32), blk(32);
    wmma_gemm<128, 2, 4, false, 0, ACT_NONE, false><<<grid, blk, 0, st>>>(C, A, B, bias, M, lda, ldb, ldc);
}
// qa: K=32, N=32 (per-head a_rel), A strided in full rows
static inline void gemm_qa(hipStream_t st, float* C, const float* A, const float* B, int M)
{
    dim3 grid(1, (M + 15) / 16), blk(32);
    wmma_gemm<32, 1, 2, false, 0, ACT_NONE, false><<<grid, blk, 0, st>>>(C, A, B, nullptr, M, EMBD, DD, EMBD);
}
// m_rel accumulate: K=32, N=32, B is K-major with ldb=32, C +=
static inline void gemm_mrel(hipStream_t st, float* C, const float* A, const float* B, int M)
{
    dim3 grid(1, (M + 15) / 16), blk(32);
    wmma_gemm<32, 1, 2, true, 32, ACT_NONE, true><<<grid, blk, 0, st>>>(C, A, B, nullptr, M, EMBD, 32, EMBD);
}

static inline void launch_fill(hipStream_t st, float* p, long n, float v)
{
    fill_kernel<<<(unsigned)((n + 255) / 256), 256, 0, st>>>(p, n, v);
}

// ---------------------------------------------------------------------------
extern "C" void kernel_launch(void* const* d_in, const int* in_sizes, int n_in,
                              void* d_out, int out_size, void* d_ws, size_t ws_size,
                              hipStream_t stream)
{
    const float* x_in[3]   = {(const float*)d_in[0], (const float*)d_in[1], (const float*)d_in[2]};
    const int*   edges     = (const int*)d_in[3];
    const float* proj_W    = (const float*)d_in[4];
    const float* proj_b    = (const float*)d_in[5];
    const float* event_emb = (const float*)d_in[6];
    const float* kqv_W     = (const float*)d_in[7];
    const float* kqv_b     = (const float*)d_in[8];
    const float* out_W     = (const float*)d_in[9];
    const float* out_b     = (const float*)d_in[10];
    const float* skip      = (const float*)d_in[11];
    const float* a_rel     = (const float*)d_in[12];
    const float* m_rel     = (const float*)d_in[13];
    const float* p_rel     = (const float*)d_in[14];
    const float* bln_g     = (const float*)d_in[15];
    const float* bln_b     = (const float*)d_in[16];
    const float* fin_W     = (const float*)d_in[17];
    const float* fin_b     = (const float*)d_in[18];
    const float* fout_W    = (const float*)d_in[19];
    const float* fout_b    = (const float*)d_in[20];
    const float* fln_g     = (const float*)d_in[21];
    const float* fln_b     = (const float*)d_in[22];
    float* outp = (float*)d_out;
    float* ws   = (float*)d_ws;

    // --- workspace layout (floats); total ~141.8M floats (~567 MB) ----------
    size_t off = 0;
    auto take = [&](size_t n) { size_t o = off; off += n; return o; };
    const size_t oXS0  = take((size_t)NT_ALL * EMBD);               // initial xs
    size_t oL[3];
    for (int l = 0; l < 3; ++l) oL[l] = take((size_t)75000 * EMBD); // layer outs t=0..2
    size_t oEV[2];
    for (int i = 0; i < 2; ++i) oEV[i] = take((size_t)50000 * EMBD);// event ping-pong
    const size_t oKB   = take((size_t)NT_ALL * EMBD);
    const size_t oQB   = take((size_t)NT_ALL * EMBD);
    const size_t oVB   = take((size_t)NT_ALL * EMBD);
    const size_t oQA   = take((size_t)50000 * EMBD);
    const size_t oTMP  = take((size_t)50000 * EMBD);                // QA+TMP contiguous (reused as ATT)
    const size_t oLOG  = take((size_t)8 * EE * HH);
    const size_t oMX   = take((size_t)NT_ALL * HH);
    const size_t oDEN  = take((size_t)NT_ALL * HH);
    const size_t oAGG  = take((size_t)NT_ALL * EMBD);
    (void)ws_size; (void)in_sizes; (void)n_in; (void)out_size;

    float* XS0 = ws + oXS0;
    float* Lb[3] = {ws + oL[0], ws + oL[1], ws + oL[2]};
    float* EVb[2] = {ws + oEV[0], ws + oEV[1]};
    float* KB = ws + oKB; float* QB = ws + oQB; float* VB = ws + oVB;
    float* QA = ws + oQA; float* TMP = ws + oTMP;
    float* LOG = ws + oLOG; float* MX = ws + oMX; float* DEN = ws + oDEN;
    float* AGG = ws + oAGG;

    // ---- stage 0: input projections + event embedding ----------------------
    for (int t = 0; t < 3; ++t)
        gemm_proj(stream, XS0 + (size_t)NODE_OFF[t] * EMBD, x_in[t],
                  proj_W + (size_t)t * EMBD * 64, proj_b + (size_t)t * EMBD, CNT[t]);
    hipMemcpyAsync(XS0 + (size_t)NODE_OFF[3] * EMBD, event_emb,
                   (size_t)50000 * EMBD * sizeof(float),
                   hipMemcpyDeviceToDevice, stream);

    // ---- 3 HGT layers -------------------------------------------------------
    for (int l = 0; l < 3; ++l) {
        const float* xin[4];
        for (int t = 0; t < 4; ++t) {
            if (l == 0)      xin[t] = XS0 + (size_t)NODE_OFF[t] * EMBD;
            else if (t < 3)  xin[t] = Lb[l - 1] + (size_t)NODE_OFF[t] * EMBD;
            else             xin[t] = EVb[(l - 1) & 1];
        }

        // K/Q/V projections (dense WMMA GEMMs)
        float* kqvDst[3] = {KB, QB, VB};
        for (int t = 0; t < 4; ++t)
            for (int k = 0; k < 3; ++k)
                gemm_k128(stream, kqvDst[k] + (size_t)NODE_OFF[t] * EMBD, xin[t],
                          kqv_W + ((((size_t)l * 4 + t) * 3 + k) * EMBD * EMBD),
                          kqv_b + (((size_t)l * 4 + t) * 3 + k) * EMBD,
                          CNT[t], EMBD, EMBD, EMBD, EMBD);

        launch_fill(stream, MX, (long)NT_ALL * HH, -INFINITY);
        launch_fill(stream, DEN, (long)NT_ALL * HH, 0.f);
        launch_fill(stream, AGG, (long)NT_ALL * EMBD, 0.f);

        // pass 1: qa = Q @ a_rel^T (WMMA), then per-edge gather-dot logits + max
        for (int r = 0; r < 8; ++r) {
            const int s = ES[r], t = ET[r];
            for (int h = 0; h < HH; ++h)
                gemm_qa(stream, QA + h * DD,
                        QB + (size_t)NODE_OFF[t] * EMBD + h * DD,
                        a_rel + (((size_t)l * 8 + r) * HH + h) * DD * DD, CNT[t]);
            logit_kernel<<<(EE * HH + 255) / 256, 256, 0, stream>>>(
                LOG + (size_t)r * EE * HH, MX, KB, QA,
                edges + (size_t)(r * 2 + 0) * EE, edges + (size_t)(r * 2 + 1) * EE,
                p_rel + ((size_t)l * 8 + r) * HH, NODE_OFF[s], NODE_OFF[t]);
        }

        // pass 2: weighted V scatter, then m_rel GEMM-accumulate into AGG
        for (int r = 0; r < 8; ++r) {
            const int s = ES[r], t = ET[r];
            launch_fill(stream, TMP, (long)CNT[t] * EMBD, 0.f);
            edge_accum_kernel<<<(unsigned)(((long)EE * EMBD + 255) / 256), 256, 0, stream>>>(
                TMP, DEN, LOG + (size_t)r * EE * HH, MX, VB,
                edges + (size_t)(r * 2 + 0) * EE, edges + (size_t)(r * 2 + 1) * EE,
                NODE_OFF[s], NODE_OFF[t]);
            for (int h = 0; h < HH; ++h)
                gemm_mrel(stream, AGG + (size_t)NODE_OFF[t] * EMBD + h * DD,
                          TMP + h * DD,
                          m_rel + (((size_t)l * 8 + r) * HH + h) * DD * DD, CNT[t]);
        }

        // per-node: normalize, gelu, out-proj (WMMA), skip-mix + residual + LN + relu
        for (int t = 0; t < 4; ++t) {
            if (l == 2 && t == 3) continue;  // event output of last layer unused
            gelu_div_kernel<<<(unsigned)(((long)CNT[t] * EMBD + 255) / 256), 256, 0, stream>>>(
                QA, AGG + (size_t)NODE_OFF[t] * EMBD, DEN + (size_t)NODE_OFF[t] * HH, CNT[t]);
            gemm_k128(stream, TMP, QA,
                      out_W + ((size_t)l * 4 + t) * EMBD * EMBD,
                      out_b + ((size_t)l * 4 + t) * EMBD,
                      CNT[t], EMBD, EMBD, EMBD, EMBD);
            float* dst = (t < 3) ? (Lb[l] + (size_t)NODE_OFF[t] * EMBD) : EVb[l & 1];
            node_fin_kernel<<<CNT[t], 128, 0, stream>>>(
                dst, TMP, xin[t], skip + (size_t)l * 4 + t,
                bln_g + (size_t)l * EMBD, bln_b + (size_t)l * EMBD);
        }
    }

    // ---- fusion (overlay scratch onto dead buffers) -------------------------
    float* SEQ  = AGG;       // <= 30000*3*128  = 11.52M  (AGG = 16M)
    float* FUSQ = KB;        // <= 90000*384    = 34.56M  (KB+QB+VB = 48M contiguous)
    float* ATT  = QA;        // <= 90000*128    = 11.52M  (QA+TMP = 12.8M contiguous)
    float* OP   = EVb[0];    // <= 90000*128    = 11.52M  (EV0+EV1 = 12.8M contiguous)

    for (int j = 0; j < 3; ++j) {
        const int N = CNT[j];
        seq_pack_kernel<<<(unsigned)(((long)N * 3 * EMBD + 255) / 256), 256, 0, stream>>>(
            SEQ, Lb[0], Lb[1], Lb[2], NODE_OFF[j], N);
        gemm_k128(stream, FUSQ, SEQ,
                  fin_W + (size_t)j * 3 * EMBD * EMBD, fin_b + (size_t)j * 3 * EMBD,
                  3 * N, 3 * EMBD, EMBD, EMBD, 3 * EMBD);
        fusion_attn_kernel<<<N, 128, 0, stream>>>(ATT, FUSQ);
        gemm_k128(stream, OP, ATT,
                  fout_W + (size_t)j * EMBD * EMBD, fout_b + (size_t)j * EMBD,
                  3 * N, EMBD, EMBD, EMBD, EMBD);
        fusion_final_kernel<<<N, 128, 0, stream>>>(
            outp + (size_t)OUT_OFF[j] * EMBD,
            OP, fln_g + (size_t)j * EMBD, fln_b + (size_t)j * EMBD);
    }
}